// _MultivariatePowerFrequencyFeatures_51256139710707
// MI455X (gfx1250) — compile-verified
//
#include <hip/hip_runtime.h>
#include <math.h>

// ---------------------------------------------------------------------------
// MultivariatePowerFrequencyFeatures for MI455X (gfx1250), fp32 end-to-end.
// Three stages, all matmul-shaped work on V_WMMA_F32_16X16X4_F32:
//   k_dft : windowed 128-pt real DFT as GEMM (frames x DFT-matrix)
//   k_csd : 16x16 cross-spectral Gram + |.|^2 + band pooling
//   k_eig : parallel Jacobi eigh + V*diag(log l)*V^T reconstruction
// ---------------------------------------------------------------------------

typedef float v2f __attribute__((ext_vector_type(2)));
typedef float v8f __attribute__((ext_vector_type(8)));

#define NFFT     128
#define HOP      32
#define T_LEN    32768
#define N_B      8
#define N_C      16
#define NFRAMES  1021          // 1 + (32768-128)/32
#define NFREQ    65
#define NWIN     252           // (1021-16)/4 + 1
#define NBAND    6
#define STFT_N   ((size_t)N_B * NFRAMES * NFREQ * N_C)   // 8,494,720 floats
#define BAND_N   ((size_t)N_B * NWIN * NBAND * 256)      // 3,096,576 floats

static __device__ __forceinline__ v8f wmma4(v2f a, v2f b, v8f c) {
  // D(16x16,f32) = A(16x4,f32) * B(4x16,f32) + C   — fp32 WMMA, wave32
  return __builtin_amdgcn_wmma_f32_16x16x4_f32(false, a, false, b,
                                               (short)0, c, false, false);
}

// ---------------------------------------------------------------------------
// Kernel 1: STFT via WMMA.
// One wave computes a 16-frame x 128-column tile for one (b,c).
// Columns 0..64  -> Re bins (cos),  65..127 -> Im bins f=1..63 (-sin).
// DFT values come from a 128-entry cosine phase table: angle = 2*pi*(k*f mod 128)/128,
// and -sin(x) = cos(x + pi/2) => phase + 32.  1/||hann|| folded into the table
// (||w||^2 = 47.625 exactly for np.hanning(128)).
// ---------------------------------------------------------------------------
__global__ __launch_bounds__(256) void k_dft(const float* __restrict__ x,
                                             float* __restrict__ sre,
                                             float* __restrict__ sim) {
  __shared__ float ct[NFFT];    // cos(2*pi*p/128) / wnorm
  __shared__ float win[NFFT];   // hann window
  const int tid = threadIdx.x;
  if (tid < NFFT) {
    const float invw = rsqrtf(47.625f);
    ct[tid]  = cosf(6.28318530717958647692f * (float)tid / 128.0f) * invw;
    win[tid] = 0.5f - 0.5f * cosf(6.28318530717958647692f * (float)tid / 127.0f);
  }
  __syncthreads();

  const int wid  = tid >> 5;
  const int lane = tid & 31;
  const int gw   = blockIdx.x * 8 + wid;      // 0 .. 8191
  const int bc   = gw >> 6;                   // (b*16 + c), 0..127
  const int tile = gw & 63;                   // 16-frame tile, 0..63
  const float* xp = x + (size_t)bc * T_LEN;

  const int m  = lane & 15;                   // A: row M / B: col N / D: col N
  const int hi = lane >> 4;                   // K sub-offset selector
  const int frame = tile * 16 + m;
  const bool fok  = frame < NFRAMES;

  v8f acc[8];
#pragma unroll
  for (int nt = 0; nt < 8; ++nt)
#pragma unroll
    for (int r = 0; r < 8; ++r) acc[nt][r] = 0.0f;

  for (int kk = 0; kk < 32; ++kk) {           // K = 128 in 32 steps of 4
    const int k0 = kk * 4 + 2 * hi;           // this lane's K pair
    v2f a;
    if (fok) {
      a = *(const v2f*)(xp + frame * HOP + k0);   // 8B aligned (k0 even)
    } else {
      a.x = 0.0f; a.y = 0.0f;
    }
    a.x *= win[k0];
    a.y *= win[k0 + 1];
#pragma unroll
    for (int nt = 0; nt < 8; ++nt) {
      const int col = nt * 16 + m;
      const int f   = (col < 65) ? col : (col - 64);
      const int pa  = (col < 65) ? 0 : 32;         // -sin via +pi/2 phase
      v2f b;
      b.x = ct[((k0       * f) + pa) & 127];
      b.y = ct[(((k0 + 1) * f) + pa) & 127];
      acc[nt] = wmma4(a, b, acc[nt]);
    }
  }

  // store: lane holds D[r + 8*hi][m] per N-tile; layout (B, frame, f, C)
  const int b = bc >> 4, c = bc & 15;
#pragma unroll
  for (int nt = 0; nt < 8; ++nt) {
    const int col = nt * 16 + m;
#pragma unroll
    for (int r = 0; r < 8; ++r) {
      const int fr = tile * 16 + r + 8 * hi;
      if (fr < NFRAMES) {
        const float v = acc[nt][r];
        if (col < 65)
          sre[(((size_t)b * NFRAMES + fr) * NFREQ + col) * N_C + c] = v;
        else
          sim[(((size_t)b * NFRAMES + fr) * NFREQ + (col - 64)) * N_C + c] = v;
      }
    }
  }
}

// ---------------------------------------------------------------------------
// Kernel 2: per (b,w): csd = X X^H, P = |csd|^2/256, band pooling.
// One wave per (b,w).  For D = A*A^T the B-fragment equals the A-fragment,
// so Re = Xr.Xr^T + Xi.Xi^T and Im = Xi.Xr^T + Xr.(-Xi)^T are WMMA chains.
// ---------------------------------------------------------------------------
__global__ __launch_bounds__(256) void k_csd(const float* __restrict__ sre,
                                             const float* __restrict__ sim,
                                             float* __restrict__ band) {
  const int tid  = threadIdx.x;
  const int wid  = tid >> 5;
  const int lane = tid & 31;
  const int gw   = blockIdx.x * 8 + wid;      // 0 .. 2015
  const int b = gw / NWIN, w = gw % NWIN;
  const int m  = lane & 15;
  const int hi = lane >> 4;

  // band boundaries over f (freq = f*15.625 Hz, mask: lo < freq <= hi)
  const int flo[NBAND] = {1, 7, 13, 20, 33, 49};
  const int fhi[NBAND] = {6, 12, 19, 32, 48, 64};
  const float fscl[NBAND] = {1.0f / (256.0f * 6.0f),  1.0f / (256.0f * 6.0f),
                             1.0f / (256.0f * 7.0f),  1.0f / (256.0f * 13.0f),
                             1.0f / (256.0f * 16.0f), 1.0f / (256.0f * 16.0f)};

  for (int k = 0; k < NBAND; ++k) {
    v8f bacc;
#pragma unroll
    for (int r = 0; r < 8; ++r) bacc[r] = 0.0f;

    for (int f = flo[k]; f <= fhi[k]; ++f) {
      v2f xr[4], xi[4], nxi[4];
#pragma unroll
      for (int kk = 0; kk < 4; ++kk) {
        const int s = kk * 4 + 2 * hi;        // K dim = frame-in-window
        const size_t i0 = (((size_t)b * NFRAMES + (4 * w + s))     * NFREQ + f) * N_C + m;
        const size_t i1 = (((size_t)b * NFRAMES + (4 * w + s + 1)) * NFREQ + f) * N_C + m;
        xr[kk].x = sre[i0]; xr[kk].y = sre[i1];
        xi[kk].x = sim[i0]; xi[kk].y = sim[i1];
        nxi[kk].x = -xi[kk].x; nxi[kk].y = -xi[kk].y;
      }
      v8f re, im;
#pragma unroll
      for (int r = 0; r < 8; ++r) { re[r] = 0.0f; im[r] = 0.0f; }
#pragma unroll
      for (int kk = 0; kk < 4; ++kk) re = wmma4(xr[kk], xr[kk], re);
#pragma unroll
      for (int kk = 0; kk < 4; ++kk) re = wmma4(xi[kk], xi[kk], re);
#pragma unroll
      for (int kk = 0; kk < 4; ++kk) im = wmma4(xi[kk], xr[kk], im);
#pragma unroll
      for (int kk = 0; kk < 4; ++kk) im = wmma4(xr[kk], nxi[kk], im);
#pragma unroll
      for (int r = 0; r < 8; ++r) bacc[r] += re[r] * re[r] + im[r] * im[r];
    }

    // store band matrix: lane element is [c = r+8*hi][d = m]
    const float sc = fscl[k];
#pragma unroll
    for (int r = 0; r < 8; ++r) {
      const int c = r + 8 * hi;
      band[((size_t)gw * NBAND + k) * 256 + c * 16 + m] = bacc[r] * sc;
    }
  }
}

// round-robin schedule: 15 rounds of 8 disjoint pairs covering all C(16,2)
static __device__ __forceinline__ void pairpq(int r, int i, int& p, int& q) {
  int a, b;
  if (i == 0) { a = 15; b = r; }
  else        { a = (r + i) % 15; b = (r + 15 - i) % 15; }
  p = a < b ? a : b;
  q = a < b ? b : a;
}

// ---------------------------------------------------------------------------
// Kernel 3: one wave per 16x16 symmetric matrix. Cyclic Jacobi in LDS,
// then out = V * diag(log lambda) * V^T via fp32 WMMA.
// ---------------------------------------------------------------------------
__global__ __launch_bounds__(256) void k_eig(const float* __restrict__ band,
                                             float* __restrict__ out) {
  __shared__ float As[8][256];
  __shared__ float Vs[8][256];
  __shared__ float Cc[8][8], Ss[8][8];
  __shared__ float Lv[8][16];

  const int tid  = threadIdx.x;
  const int wid  = tid >> 5;
  const int lane = tid & 31;
  const size_t mid = (size_t)blockIdx.x * 8 + wid;   // 0 .. 12095

#pragma unroll
  for (int i = 0; i < 8; ++i) {
    const int idx = lane + 32 * i;
    As[wid][idx] = band[mid * 256 + idx];
    Vs[wid][idx] = ((idx >> 4) == (idx & 15)) ? 1.0f : 0.0f;
  }
  __syncthreads();

  for (int sweep = 0; sweep < 10; ++sweep) {
    for (int r = 0; r < 15; ++r) {
      if (lane < 8) {                         // rotation angles, one pair/lane
        int p, q; pairpq(r, lane, p, q);
        const float app = As[wid][p * 17];
        const float aqq = As[wid][q * 17];
        const float apq = As[wid][p * 16 + q];
        float c = 1.0f, s = 0.0f;
        if (fabsf(apq) > 1e-30f) {
          const float tau = (aqq - app) / (2.0f * apq);
          float t = 1.0f / (fabsf(tau) + sqrtf(1.0f + tau * tau));
          if (tau < 0.0f) t = -t;
          c = rsqrtf(1.0f + t * t);
          s = t * c;
        }
        Cc[wid][lane] = c; Ss[wid][lane] = s;
      }
      __syncthreads();
      // A <- G^T A  (rows p,q), 8 pairs x 16 cols = 128 tasks
#pragma unroll
      for (int t2 = 0; t2 < 4; ++t2) {
        const int task = lane + 32 * t2;
        const int i = task >> 4, j = task & 15;
        int p, q; pairpq(r, i, p, q);
        const float c = Cc[wid][i], s = Ss[wid][i];
        const float apj = As[wid][p * 16 + j], aqj = As[wid][q * 16 + j];
        As[wid][p * 16 + j] = c * apj - s * aqj;
        As[wid][q * 16 + j] = s * apj + c * aqj;
      }
      __syncthreads();
      // A <- A G (cols p,q) and V <- V G
#pragma unroll
      for (int t2 = 0; t2 < 4; ++t2) {
        const int task = lane + 32 * t2;
        const int i = task >> 4, j = task & 15;
        int p, q; pairpq(r, i, p, q);
        const float c = Cc[wid][i], s = Ss[wid][i];
        const float ajp = As[wid][j * 16 + p], ajq = As[wid][j * 16 + q];
        As[wid][j * 16 + p] = c * ajp - s * ajq;
        As[wid][j * 16 + q] = s * ajp + c * ajq;
        const float vjp = Vs[wid][j * 16 + p], vjq = Vs[wid][j * 16 + q];
        Vs[wid][j * 16 + p] = c * vjp - s * vjq;
        Vs[wid][j * 16 + q] = s * vjp + c * vjq;
      }
      __syncthreads();
    }
  }

  if (lane < 16) {                            // log eigenvalues (diag)
    float lv = logf(As[wid][lane * 17]);
    if (lv != lv || lv == -__builtin_inff()) lv = 0.0f;   // nan/neginf -> 0
    Lv[wid][lane] = lv;
  }
  __syncthreads();

  // out = W * V^T,  W[m][e] = V[m][e]*lv[e];  B-frag of V^T == A-frag of V
  const int m  = lane & 15;
  const int hi = lane >> 4;
  v8f acc;
#pragma unroll
  for (int r = 0; r < 8; ++r) acc[r] = 0.0f;
#pragma unroll
  for (int kk = 0; kk < 4; ++kk) {
    const int k0 = kk * 4 + 2 * hi;
    v2f a, bb;
    bb.x = Vs[wid][m * 16 + k0];
    bb.y = Vs[wid][m * 16 + k0 + 1];
    a.x  = bb.x * Lv[wid][k0];
    a.y  = bb.y * Lv[wid][k0 + 1];
    acc = wmma4(a, bb, acc);
  }

  // output (B, band, i, j, w):  mid = ((b*252)+w)*6 + k
  const int b = (int)(mid / (NWIN * NBAND));
  const int rem = (int)(mid % (NWIN * NBAND));
  const int w = rem / NBAND;
  const int k = rem % NBAND;
#pragma unroll
  for (int r = 0; r < 8; ++r) {
    const int i = r + 8 * hi, j = m;
    out[((((size_t)b * NBAND + k) * 16 + i) * 16 + j) * NWIN + w] = acc[r];
  }
}

// ---------------------------------------------------------------------------
extern "C" void kernel_launch(void* const* d_in, const int* in_sizes, int n_in,
                              void* d_out, int out_size, void* d_ws, size_t ws_size,
                              hipStream_t stream) {
  const float* x = (const float*)d_in[0];           // (8,16,32768) fp32
  float* sre  = (float*)d_ws;                       // (8,1021,65,16)
  float* sim  = sre + STFT_N;                       // (8,1021,65,16)
  float* band = sim + STFT_N;                       // (8,252,6,16,16)

  // Im of DC (f=0) and Nyquist (f=64) bins are never written by k_dft
  hipMemsetAsync(sim, 0, STFT_N * sizeof(float), stream);

  k_dft<<<1024, 256, 0, stream>>>(x, sre, sim);     // 8192 waves: 128 bc x 64 tiles
  k_csd<<<252, 256, 0, stream>>>(sre, sim, band);   // 2016 waves: 8 b x 252 w
  k_eig<<<1512, 256, 0, stream>>>(band, (float*)d_out); // 12096 waves: 1/matrix
}